// GCN_M_91319594648156
// MI455X (gfx1250) — compile-verified
//
#include <hip/hip_runtime.h>
#include <hip/hip_bf16.h>

typedef __attribute__((ext_vector_type(2))) float v2f;
typedef __attribute__((ext_vector_type(8))) float v8f;

#define N_NODES_C 100000

// ---------------------------------------------------------------------------
// Degree / normalization kernels
// ---------------------------------------------------------------------------
__global__ void zero_f32_kernel(float* __restrict__ p, int n) {
    int i = blockIdx.x * blockDim.x + threadIdx.x;
    if (i < n) p[i] = 0.0f;
}

__global__ void deg_accum_kernel(const int* __restrict__ ei, float* __restrict__ deg, int E) {
    int e = blockIdx.x * blockDim.x + threadIdx.x;
    if (e < E) atomicAdd(&deg[ei[E + e]], 1.0f);
}

__global__ void finalize_dinv_kernel(float* __restrict__ deg, int n) {
    int i = blockIdx.x * blockDim.x + threadIdx.x;
    if (i < n) deg[i] = rsqrtf(deg[i] + 1.0f);
}

// ---------------------------------------------------------------------------
// WMMA fp32 GEMM: C[M,N] = A[M,K] @ B[K,N]
// One wave computes a 16x64 strip via 4 accumulators of V_WMMA_F32_16X16X4_F32.
// A-frag (16x4 f32): lanes 0-15 -> row M=l16, K = {k,k+1}; lanes 16-31 -> K = {k+2,k+3}
// B-frag (4x16 f32): lanes 0-15 -> col N=l16, K = {k,k+1}; lanes 16-31 -> K = {k+2,k+3}
// C/D (16x16 f32):   VGPR i -> row m0+i (lanes 0-15) / m0+8+i (lanes 16-31), col = n + l16
//
// Out-of-range rows are CLAMPED (not zeroed): D row m depends only on A row m,
// and rows >= M are never stored, so clamping is safe and keeps the inner loop
// branch-free (EXEC stays all-1s, as WMMA requires).
// ---------------------------------------------------------------------------
__global__ __launch_bounds__(256)
void gemm_wmma_f32_kernel(const float* __restrict__ A, const float* __restrict__ B,
                          float* __restrict__ C, int M, int K, int N) {
    const int lane     = threadIdx.x & 31;
    const int wave     = threadIdx.x >> 5;
    const int strips   = N >> 6;          // N/64 column strips (4 for N=256, 2 for N=128)
    const int rowTiles = 8 / strips;      // row tiles per block
    const int strip    = wave % strips;
    const int rowTile  = wave / strips;
    const int m0       = blockIdx.x * (rowTiles * 16) + rowTile * 16;
    const int n0       = strip * 64;
    const int l16      = lane & 15;
    const int hi       = lane >> 4;       // 0 or 1: selects K pair {0,1} vs {2,3}

    if (m0 >= M) return;                  // wave-uniform exit

    const int arow = min(m0 + l16, M - 1);          // clamp: branch-free A loads
    const float* Arow = A + (size_t)arow * K + 2 * hi;

    v8f c0 = {}, c1 = {}, c2 = {}, c3 = {};

    for (int k = 0; k < K; k += 4) {
        v2f a = *(const v2f*)(Arow + k);            // 8B-aligned -> global_load_b64

        const float* Bk0 = B + (size_t)(k + 2 * hi) * N + n0 + l16;
        const float* Bk1 = Bk0 + N;
        v2f b0, b1, b2, b3;
        b0.x = Bk0[0];  b0.y = Bk1[0];
        b1.x = Bk0[16]; b1.y = Bk1[16];
        b2.x = Bk0[32]; b2.y = Bk1[32];
        b3.x = Bk0[48]; b3.y = Bk1[48];

        c0 = __builtin_amdgcn_wmma_f32_16x16x4_f32(false, a, false, b0, (short)0, c0, false, false);
        c1 = __builtin_amdgcn_wmma_f32_16x16x4_f32(false, a, false, b1, (short)0, c1, false, false);
        c2 = __builtin_amdgcn_wmma_f32_16x16x4_f32(false, a, false, b2, (short)0, c2, false, false);
        c3 = __builtin_amdgcn_wmma_f32_16x16x4_f32(false, a, false, b3, (short)0, c3, false, false);
    }

    const int rbase = m0 + hi * 8;
    for (int i = 0; i < 8; ++i) {
        int r = rbase + i;
        if (r < M) {
            float* Crow = C + (size_t)r * N + n0 + l16;
            Crow[0]  = c0[i];
            Crow[16] = c1[i];
            Crow[32] = c2[i];
            Crow[48] = c3[i];
        }
    }
}

// ---------------------------------------------------------------------------
// agg[i,f] = hlin[i,f] * dinv[i]^2 + b[f]   (self-loop term + bias, also zero-init)
// ---------------------------------------------------------------------------
__global__ void init_agg_kernel(const float* __restrict__ hlin, const float* __restrict__ dinv,
                                const float* __restrict__ b, float* __restrict__ agg,
                                int Nn, int F) {
    int i = blockIdx.x * blockDim.x + threadIdx.x;
    int total = Nn * F;
    if (i >= total) return;
    int node = i / F;
    int f    = i - node * F;
    float di = dinv[node];
    agg[i] = hlin[i] * di * di + b[f];
}

// ---------------------------------------------------------------------------
// One wave per edge: agg[dst,:] += hlin[src,:] * (dinv[src]*dinv[dst])
// ---------------------------------------------------------------------------
__global__ __launch_bounds__(256)
void scatter_edges_kernel(const float* __restrict__ hlin, const int* __restrict__ ei,
                          const float* __restrict__ dinv, float* __restrict__ agg,
                          int E, int F) {
    const int lane = threadIdx.x & 31;
    const int e    = (blockIdx.x * blockDim.x + threadIdx.x) >> 5;
    if (e >= E) return;
    const int s = ei[e];
    const int d = ei[E + e];
    const float norm = dinv[s] * dinv[d];
    const float4* srcRow = (const float4*)(hlin + (size_t)s * F);
    float* dstRow = agg + (size_t)d * F;
    const int nvec = F >> 2;
    for (int v = lane; v < nvec; v += 32) {
        float4 m = srcRow[v];
        int f = v * 4;
        atomicAdd(dstRow + f + 0, m.x * norm);
        atomicAdd(dstRow + f + 1, m.y * norm);
        atomicAdd(dstRow + f + 2, m.z * norm);
        atomicAdd(dstRow + f + 3, m.w * norm);
    }
}

// ---------------------------------------------------------------------------
// One wave per label edge: res[l] = sum_f out[s,f]*out[d,f]*Wp[f] + bp
// ---------------------------------------------------------------------------
__global__ __launch_bounds__(256)
void predict_kernel(const float* __restrict__ feat, const int* __restrict__ eli,
                    const float* __restrict__ Wp, const float* __restrict__ bp,
                    float* __restrict__ res, int L) {
    const int lane = threadIdx.x & 31;
    const int l    = (blockIdx.x * blockDim.x + threadIdx.x) >> 5;
    if (l >= L) return;
    const int s = eli[l];
    const int d = eli[L + l];
    const float4* sr = (const float4*)(feat + (size_t)s * 128);
    const float4* dr = (const float4*)(feat + (size_t)d * 128);
    const float4* wp = (const float4*)Wp;
    float4 a = sr[lane];
    float4 b = dr[lane];
    float4 w = wp[lane];
    float acc = a.x * b.x * w.x + a.y * b.y * w.y + a.z * b.z * w.z + a.w * b.w * w.w;
    #pragma unroll
    for (int off = 16; off; off >>= 1) acc += __shfl_xor(acc, off, 32);
    if (lane == 0) res[l] = acc + bp[0];
}

// ---------------------------------------------------------------------------
// Host side
// ---------------------------------------------------------------------------
static inline void launch_gemm(const float* A, const float* B, float* C,
                               int M, int K, int N, hipStream_t s) {
    int strips = N / 64;
    int rowsPerBlock = (8 / strips) * 16;
    int grid = (M + rowsPerBlock - 1) / rowsPerBlock;
    gemm_wmma_f32_kernel<<<grid, 256, 0, s>>>(A, B, C, M, K, N);
}

extern "C" void kernel_launch(void* const* d_in, const int* in_sizes, int n_in,
                              void* d_out, int out_size, void* d_ws, size_t ws_size,
                              hipStream_t stream) {
    const float* x   = (const float*)d_in[0];
    const int*   ei  = (const int*)d_in[1];
    const int*   eli = (const int*)d_in[2];
    const float* W1  = (const float*)d_in[3];
    const float* b1  = (const float*)d_in[4];
    const float* W2  = (const float*)d_in[5];
    const float* b2  = (const float*)d_in[6];
    const float* W3  = (const float*)d_in[7];
    const float* b3  = (const float*)d_in[8];
    const float* Wp  = (const float*)d_in[9];
    const float* bp  = (const float*)d_in[10];
    float* out = (float*)d_out;

    const int Nn = N_NODES_C;
    const int E  = in_sizes[1] / 2;
    const int L  = in_sizes[2] / 2;
    const int H  = 256, F0 = 128;

    // Workspace layout (floats): dinv | P (GEMM out) | Q | R  (node feature buffers)
    float* dinv = (float*)d_ws;
    float* P = dinv + 131072;                   // 512KB pad after dinv
    float* Q = P + (size_t)Nn * H;
    float* R = Q + (size_t)Nn * H;

    const int T = 256;

    // Degree + symmetric normalization
    zero_f32_kernel<<<(Nn + T - 1) / T, T, 0, stream>>>(dinv, Nn);
    deg_accum_kernel<<<(E + T - 1) / T, T, 0, stream>>>(ei, dinv, E);
    finalize_dinv_kernel<<<(Nn + T - 1) / T, T, 0, stream>>>(dinv, Nn);

    // Layer 1: x[N,128] @ W1[128,256] -> P; agg -> Q
    launch_gemm(x, W1, P, Nn, F0, H, stream);
    init_agg_kernel<<<(Nn * H + T - 1) / T, T, 0, stream>>>(P, dinv, b1, Q, Nn, H);
    scatter_edges_kernel<<<(E * 32 + T - 1) / T, T, 0, stream>>>(P, ei, dinv, Q, E, H);

    // Layer 2: Q[N,256] @ W2[256,256] -> P; agg -> R
    launch_gemm(Q, W2, P, Nn, H, H, stream);
    init_agg_kernel<<<(Nn * H + T - 1) / T, T, 0, stream>>>(P, dinv, b2, R, Nn, H);
    scatter_edges_kernel<<<(E * 32 + T - 1) / T, T, 0, stream>>>(P, ei, dinv, R, E, H);

    // Layer 3: R[N,256] @ W3[256,128] -> P; agg -> Q
    launch_gemm(R, W3, P, Nn, H, F0, stream);
    init_agg_kernel<<<(Nn * F0 + T - 1) / T, T, 0, stream>>>(P, dinv, b3, Q, Nn, F0);
    scatter_edges_kernel<<<(E * 32 + T - 1) / T, T, 0, stream>>>(P, ei, dinv, Q, E, F0);

    // Link prediction
    predict_kernel<<<(L * 32 + T - 1) / T, T, 0, stream>>>(Q, eli, Wp, bp, out, L);
}